// AttnDecoder_9672266350594
// MI455X (gfx1250) — compile-verified
//
#include <hip/hip_runtime.h>
#include <hip/hip_bf16.h>
#include <math.h>

// ---------------------------------------------------------------------------
// Problem constants
// ---------------------------------------------------------------------------
#define VOC   50000
#define HD    256          // hidden dim
#define H2    512          // lstm hidden (2*H)
#define BB    128          // batch
#define TK    400
#define NOOV  100
#define VEXT  (VOC + NOOV) // 50100

typedef float v2f __attribute__((ext_vector_type(2)));
typedef float v8f __attribute__((ext_vector_type(8)));

// V_WMMA_F32_16X16X4_F32: D(16x16 f32) = A(16x4 f32) * B(4x16 f32) + C
// A/B: 2 VGPRs each; lanes 0-15 hold K={k0,k0+1}, lanes 16-31 hold K={k0+2,k0+3}
// C/D: 8 VGPRs; element (r,lane) = [M = r + 8*(lane>=16), N = lane%16]
__device__ __forceinline__ v8f wmma4(v2f a, v2f b, v8f c) {
    return __builtin_amdgcn_wmma_f32_16x16x4_f32(
        /*neg_a=*/false, a, /*neg_b=*/false, b,
        /*c_mod=*/(short)0, c, /*reuse_a=*/false, /*reuse_b=*/false);
}

__device__ __forceinline__ float sigmoidf_(float x) {
    return 1.0f / (1.0f + expf(-x));
}

// CDNA5 async copy: GLOBAL_LOAD_ASYNC_TO_LDS_B128 (tracked by ASYNCcnt).
// VDST = 32-bit LDS byte address, VADDR = 64-bit global address.
// Generic shared-pointer low 32 bits == LDS byte offset (aperture in high bits).
__device__ __forceinline__ void async_load_b128(uint32_t lds_addr, const void* gptr) {
    asm volatile("global_load_async_to_lds_b128 %0, %1, off"
                 :: "v"(lds_addr), "v"(gptr) : "memory");
}
__device__ __forceinline__ void wait_asynccnt0() {
    asm volatile("s_wait_asynccnt 0" ::: "memory");
}

// ---------------------------------------------------------------------------
// 1) embedding gather: emb_x[b][j] = emb[input[b]][j]
// ---------------------------------------------------------------------------
__global__ __launch_bounds__(HD)
void embed_kernel(const int* __restrict__ input, const float* __restrict__ emb,
                  float* __restrict__ emb_x) {
    int b = blockIdx.x;
    int j = threadIdx.x;
    emb_x[b * HD + j] = emb[(size_t)input[b] * HD + j];
}

// ---------------------------------------------------------------------------
// 2) gates = emb_x @ w_ih.T + h_prev @ w_hh.T + b_ih + b_hh   [128 x 2048]
//    one wave per 16x16 output tile, fp32 WMMA, K=4 per op
// ---------------------------------------------------------------------------
__global__ __launch_bounds__(32)
void gates_kernel(const float* __restrict__ emb_x,   // [128,256]
                  const float* __restrict__ h_prev,  // [128,512]
                  const float* __restrict__ w_ih,    // [2048,256]
                  const float* __restrict__ w_hh,    // [2048,512]
                  const float* __restrict__ b_ih,    // [2048]
                  const float* __restrict__ b_hh,    // [2048]
                  float* __restrict__ gates)         // [128,2048]
{
    const int n0   = blockIdx.x * 16;
    const int m0   = blockIdx.y * 16;
    const int lane = threadIdx.x;
    const int half = lane >> 4;     // 0: K pair {0,1}, 1: K pair {2,3}
    const int lm   = lane & 15;     // A row (m) / B col (n)

    v8f c = {};
    {   // x @ w_ih.T  (K = 256)
        const float* arow = emb_x + (m0 + lm) * HD;
        const float* brow = w_ih  + (size_t)(n0 + lm) * HD;
        for (int k0 = 0; k0 < HD; k0 += 4) {
            int ka = k0 + 2 * half;
            v2f a = { arow[ka], arow[ka + 1] };
            v2f b = { brow[ka], brow[ka + 1] };
            c = wmma4(a, b, c);
        }
    }
    {   // h_prev @ w_hh.T  (K = 512)
        const float* arow = h_prev + (m0 + lm) * H2;
        const float* brow = w_hh   + (size_t)(n0 + lm) * H2;
        for (int k0 = 0; k0 < H2; k0 += 4) {
            int ka = k0 + 2 * half;
            v2f a = { arow[ka], arow[ka + 1] };
            v2f b = { brow[ka], brow[ka + 1] };
            c = wmma4(a, b, c);
        }
    }
    float bias = b_ih[n0 + lm] + b_hh[n0 + lm];
    for (int r = 0; r < 8; ++r) {
        int m = m0 + r + 8 * half;
        gates[m * (4 * H2) + n0 + lm] = c[r] + bias;
    }
}

// ---------------------------------------------------------------------------
// 3) LSTM pointwise: i,f,g,o -> c_new, h_new (written into d_out tail)
// ---------------------------------------------------------------------------
__global__ __launch_bounds__(256)
void lstm_pw_kernel(const float* __restrict__ gates, const float* __restrict__ c_prev,
                    float* __restrict__ h_new, float* __restrict__ c_new) {
    int i = blockIdx.x * blockDim.x + threadIdx.x;
    if (i >= BB * H2) return;
    int b = i >> 9, j = i & (H2 - 1);
    const float* g = gates + b * (4 * H2);
    float ig = sigmoidf_(g[j]);
    float fg = sigmoidf_(g[H2 + j]);
    float gg = tanhf(g[2 * H2 + j]);
    float og = sigmoidf_(g[3 * H2 + j]);
    float cn = fg * c_prev[i] + ig * gg;
    c_new[i] = cn;
    h_new[i] = og * tanhf(cn);
}

// ---------------------------------------------------------------------------
// 4) hW[b][n] = sum_k h_new[b][k] * w_att1[n][k] + b_att1[n]   (h half of W1)
// ---------------------------------------------------------------------------
__global__ __launch_bounds__(H2)
void hW_kernel(const float* __restrict__ h_new, const float* __restrict__ w_att1,
               const float* __restrict__ b_att1, float* __restrict__ hW) {
    __shared__ float hrow[H2];
    int b = blockIdx.x;
    hrow[threadIdx.x] = h_new[b * H2 + threadIdx.x];
    __syncthreads();
    int n = threadIdx.x;
    const float* wr = w_att1 + (size_t)n * (2 * H2);   // first 512 cols = h part
    float s = 0.0f;
    for (int k = 0; k < H2; ++k) s += hrow[k] * wr[k];
    hW[b * H2 + n] = s + b_att1[n];
}

// ---------------------------------------------------------------------------
// 5) attention scorer: e[b][t] = w_att2 . tanh(enc[b,t,:]@W1_e.T + hW[b,:])
//    one wave per (batch, 16-row t-tile); WMMA over K=512, fused tanh+dot
// ---------------------------------------------------------------------------
__global__ __launch_bounds__(32)
void att_score_kernel(const float* __restrict__ enc,    // [128,400,512]
                      const float* __restrict__ w_att1, // [512,1024]
                      const float* __restrict__ hW,     // [128,512]
                      const float* __restrict__ w_att2, // [512]
                      float* __restrict__ e)            // [128,400]
{
    const int b    = blockIdx.y;
    const int t0   = blockIdx.x * 16;
    const int lane = threadIdx.x;
    const int half = lane >> 4;
    const int lm   = lane & 15;

    float esum[8];
    for (int r = 0; r < 8; ++r) esum[r] = 0.0f;

    const float* A = enc + (size_t)b * TK * H2 + (size_t)(t0 + lm) * H2;
    for (int n0 = 0; n0 < H2; n0 += 16) {
        v8f c = {};
        const float* brow = w_att1 + (size_t)(n0 + lm) * (2 * H2) + H2; // enc half
        for (int k0 = 0; k0 < H2; k0 += 4) {
            int ka = k0 + 2 * half;
            v2f a  = { A[ka], A[ka + 1] };
            v2f bb = { brow[ka], brow[ka + 1] };
            c = wmma4(a, bb, c);
        }
        float hw = hW[b * H2 + n0 + lm];     // depends on n only
        float wa = w_att2[n0 + lm];
        for (int r = 0; r < 8; ++r)
            esum[r] += tanhf(c[r] + hw) * wa;
    }
    // reduce over the 16 lanes of each half (each half owns rows r / r+8)
    for (int r = 0; r < 8; ++r) {
        float v = esum[r];
        for (int off = 1; off < 16; off <<= 1)
            v += __shfl_xor(v, off, 32);
        esum[r] = v;
    }
    if (lm == 0) {
        for (int r = 0; r < 8; ++r)
            e[b * TK + t0 + r + 8 * half] = esum[r];
    }
}

// ---------------------------------------------------------------------------
// 6) softmax over TK (in place: e -> a)
// ---------------------------------------------------------------------------
__global__ __launch_bounds__(512)
void att_softmax_kernel(float* __restrict__ e) {
    __shared__ float buf[512];
    __shared__ float red[512];
    int b = blockIdx.x;
    float v = (threadIdx.x < TK) ? e[b * TK + threadIdx.x] : -INFINITY;
    buf[threadIdx.x] = v;
    red[threadIdx.x] = v;
    __syncthreads();
    for (int s = 256; s > 0; s >>= 1) {
        if (threadIdx.x < s) red[threadIdx.x] = fmaxf(red[threadIdx.x], red[threadIdx.x + s]);
        __syncthreads();
    }
    float m = red[0];
    __syncthreads();
    float ex = (threadIdx.x < TK) ? expf(buf[threadIdx.x] - m) : 0.0f;
    red[threadIdx.x] = ex;
    __syncthreads();
    for (int s = 256; s > 0; s >>= 1) {
        if (threadIdx.x < s) red[threadIdx.x] += red[threadIdx.x + s];
        __syncthreads();
    }
    if (threadIdx.x < TK) e[b * TK + threadIdx.x] = ex / red[0];
}

// ---------------------------------------------------------------------------
// 7) context[b][n] = sum_t a[b][t] * enc[b][t][n]
// ---------------------------------------------------------------------------
__global__ __launch_bounds__(H2)
void context_kernel(const float* __restrict__ a, const float* __restrict__ enc,
                    float* __restrict__ ctx) {
    __shared__ float arow[TK];
    int b = blockIdx.x;
    if (threadIdx.x < TK) arow[threadIdx.x] = a[b * TK + threadIdx.x];
    __syncthreads();
    int n = threadIdx.x;
    const float* er = enc + (size_t)b * TK * H2 + n;
    float s = 0.0f;
    for (int t = 0; t < TK; ++t) s += arow[t] * er[(size_t)t * H2];
    ctx[b * H2 + n] = s;
}

// ---------------------------------------------------------------------------
// 8) s_cat = [h_new, context]  [128 x 1024]
// ---------------------------------------------------------------------------
__global__ __launch_bounds__(256)
void scat_build_kernel(const float* __restrict__ h_new, const float* __restrict__ ctx,
                       float* __restrict__ s_cat) {
    int i = blockIdx.x * blockDim.x + threadIdx.x;
    if (i >= BB * 2 * H2) return;
    int b = i >> 10, j = i & (2 * H2 - 1);
    s_cat[i] = (j < H2) ? h_new[b * H2 + j] : ctx[b * H2 + (j - H2)];
}

// ---------------------------------------------------------------------------
// 9) p_gen[b] = clip(sigmoid([ctx,h,emb] . w_gen + b_gen), 1e-8)
// ---------------------------------------------------------------------------
__global__ __launch_bounds__(256)
void pgen_kernel(const float* __restrict__ ctx, const float* __restrict__ h_new,
                 const float* __restrict__ emb_x, const float* __restrict__ w_gen,
                 const float* __restrict__ b_gen, float* __restrict__ pgen) {
    __shared__ float red[256];
    int b = blockIdx.x;
    float s = 0.0f;
    for (int j = threadIdx.x; j < 5 * HD; j += 256) {
        float x = (j < H2)     ? ctx[b * H2 + j]
                : (j < 2 * H2) ? h_new[b * H2 + (j - H2)]
                               : emb_x[b * HD + (j - 2 * H2)];
        s += x * w_gen[j];
    }
    red[threadIdx.x] = s;
    __syncthreads();
    for (int t = 128; t > 0; t >>= 1) {
        if (threadIdx.x < t) red[threadIdx.x] += red[threadIdx.x + t];
        __syncthreads();
    }
    if (threadIdx.x == 0)
        pgen[b] = fmaxf(sigmoidf_(red[0] + b_gen[0]), 1e-8f);
}

// ---------------------------------------------------------------------------
// 10) FC logits: [128 x 50000] = s_cat @ w_fc.T + b_fc, staged into d_out.
//     Block = 8 waves = all 8 M-tiles, so w_fc streams from HBM once.
//     w_fc tiles are double-buffered in LDS via GLOBAL_LOAD_ASYNC_TO_LDS_B128
//     (ASYNCcnt), overlapping the HBM stream with the WMMA K-loop.
//     LDS row stride 68 floats: 16B-aligned B128 writes, bank-conflict-free
//     reads (bank = 4n + k mod 64).
// ---------------------------------------------------------------------------
__global__ __launch_bounds__(256)
void fc_kernel(const float* __restrict__ s_cat, // [128,1024]
               const float* __restrict__ w_fc,  // [50000,1024]
               const float* __restrict__ b_fc,  // [50000]
               float* __restrict__ out)         // [128,50100], logits in [:,0:50000]
{
    __shared__ float bT[2][16 * 68];
    const int n0   = blockIdx.x * 16;
    const int tid  = threadIdx.x;
    const int wave = tid >> 5;     // m-tile 0..7
    const int lane = tid & 31;
    const int half = lane >> 4;
    const int lm   = lane & 15;

    // per-thread async-copy addressing: 256 threads x 16B = one 16x64 f32 tile
    const int ln = tid >> 4;          // w_fc row within tile: 0..15
    const int lk = (tid & 15) * 4;    // k offset within chunk: 0,4,...,60
    const float* gsrc = w_fc + (size_t)(n0 + ln) * (2 * H2) + lk;
    const uint32_t lds0 = (uint32_t)(uintptr_t)&bT[0][ln * 68 + lk];
    const uint32_t lds1 = (uint32_t)(uintptr_t)&bT[1][ln * 68 + lk];

    async_load_b128(lds0, gsrc);      // prime buffer 0 (kc = 0)

    v8f c = {};
    const float* Arow = s_cat + (size_t)(wave * 16 + lm) * (2 * H2);
    for (int kc = 0; kc < 16; ++kc) {
        wait_asynccnt0();             // my slice of buf[kc&1] landed
        __syncthreads();              // everyone's slice landed
        if (kc + 1 < 16)
            async_load_b128((kc & 1) ? lds0 : lds1,
                            gsrc + (size_t)(kc + 1) * 64);
        const float* bbuf = bT[kc & 1];
        const float* Ak   = Arow + kc * 64;
        for (int k0 = 0; k0 < 64; k0 += 4) {
            int ka = k0 + 2 * half;
            v2f a  = { Ak[ka], Ak[ka + 1] };
            v2f bb = { bbuf[lm * 68 + ka], bbuf[lm * 68 + ka + 1] };
            c = wmma4(a, bb, c);
        }
        // no trailing barrier needed: next overwrite of bT[kc&1] is issued in
        // iteration kc+1, strictly after the wait+barrier at top of kc+1.
    }
    float bias = b_fc[n0 + lm];
    for (int r = 0; r < 8; ++r) {
        int m = wave * 16 + r + 8 * half;
        out[(size_t)m * VEXT + n0 + lm] = c[r] + bias;
    }
}

// ---------------------------------------------------------------------------
// 11) per-row max & sum(exp) over 50000 logits
// ---------------------------------------------------------------------------
__global__ __launch_bounds__(256)
void rowstat_kernel(const float* __restrict__ out, float* __restrict__ rmax,
                    float* __restrict__ rsum) {
    __shared__ float red[256];
    int b = blockIdx.x;
    const float* row = out + (size_t)b * VEXT;
    float m = -INFINITY;
    for (int v = threadIdx.x; v < VOC; v += 256) m = fmaxf(m, row[v]);
    red[threadIdx.x] = m;
    __syncthreads();
    for (int s = 128; s > 0; s >>= 1) {
        if (threadIdx.x < s) red[threadIdx.x] = fmaxf(red[threadIdx.x], red[threadIdx.x + s]);
        __syncthreads();
    }
    m = red[0];
    __syncthreads();
    float sum = 0.0f;
    for (int v = threadIdx.x; v < VOC; v += 256) sum += expf(row[v] - m);
    red[threadIdx.x] = sum;
    __syncthreads();
    for (int s = 128; s > 0; s >>= 1) {
        if (threadIdx.x < s) red[threadIdx.x] += red[threadIdx.x + s];
        __syncthreads();
    }
    if (threadIdx.x == 0) { rmax[b] = m; rsum[b] = red[0]; }
}

// ---------------------------------------------------------------------------
// 12) p_vocab_cat: in-place logits -> softmax * p_gen ; OOV extension = 0
// ---------------------------------------------------------------------------
__global__ __launch_bounds__(256)
void pvocab_kernel(float* __restrict__ out, const float* __restrict__ rmax,
                   const float* __restrict__ rsum, const float* __restrict__ pgen) {
    size_t i = (size_t)blockIdx.x * blockDim.x + threadIdx.x;
    if (i >= (size_t)BB * VEXT) return;
    int b = (int)(i / VEXT);
    int v = (int)(i % VEXT);
    if (v < VOC)
        out[i] = expf(out[i] - rmax[b]) / rsum[b] * pgen[b];
    else
        out[i] = 0.0f;
}

// ---------------------------------------------------------------------------
// 13) scatter-add (1-p_gen)*a_t at story indices; sequential per row
//     (deterministic; handles repeated indices exactly like the reference)
// ---------------------------------------------------------------------------
__global__ __launch_bounds__(128)
void scatter_kernel(float* __restrict__ out, const int* __restrict__ story,
                    const float* __restrict__ a, const float* __restrict__ pgen) {
    int b = blockIdx.x * blockDim.x + threadIdx.x;
    if (b >= BB) return;
    float s = 1.0f - pgen[b];
    float* row = out + (size_t)b * VEXT;
    for (int t = 0; t < TK; ++t)
        row[story[b * TK + t]] += s * a[b * TK + t];
}

// ---------------------------------------------------------------------------
// 14) out = log(max(out, 1e-8))
// ---------------------------------------------------------------------------
__global__ __launch_bounds__(256)
void log_kernel(float* __restrict__ out) {
    size_t i = (size_t)blockIdx.x * blockDim.x + threadIdx.x;
    if (i >= (size_t)BB * VEXT) return;
    out[i] = logf(fmaxf(out[i], 1e-8f));
}

// ---------------------------------------------------------------------------
// launch
// ---------------------------------------------------------------------------
extern "C" void kernel_launch(void* const* d_in, const int* in_sizes, int n_in,
                              void* d_out, int out_size, void* d_ws, size_t ws_size,
                              hipStream_t stream) {
    const int*   input  = (const int*)  d_in[0];
    const float* dh     = (const float*)d_in[1];   // h_prev [1,128,512]
    const float* cell   = (const float*)d_in[2];   // c_prev
    const float* enc    = (const float*)d_in[3];   // [128,400,512]
    const int*   story  = (const int*)  d_in[4];   // [128,400]
    /* d_in[5] extra_zeros unused: zeros written directly */
    const float* emb    = (const float*)d_in[6];
    const float* w_ih   = (const float*)d_in[7];
    const float* w_hh   = (const float*)d_in[8];
    const float* b_ih   = (const float*)d_in[9];
    const float* b_hh   = (const float*)d_in[10];
    const float* w_att1 = (const float*)d_in[11];
    const float* b_att1 = (const float*)d_in[12];
    const float* w_att2 = (const float*)d_in[13];
    const float* w_fc   = (const float*)d_in[14];
    const float* b_fc   = (const float*)d_in[15];
    const float* w_gen  = (const float*)d_in[16];
    const float* b_gen  = (const float*)d_in[17];

    float* out   = (float*)d_out;                     // [128,50100] log-probs
    float* h_new = out + (size_t)BB * VEXT;           // [1,128,512]
    float* c_new = h_new + (size_t)BB * H2;           // [1,128,512]

    float* ws    = (float*)d_ws;
    float* gates = ws;                 // 128*2048   = 262144
    float* emb_x = ws + 262144;        // 128*256    = 32768
    float* hW    = ws + 294912;        // 128*512    = 65536
    float* e_a   = ws + 360448;        // 128*400    = 51200 (e then a, in place)
    float* ctx   = ws + 411648;        // 128*512    = 65536
    float* pgen  = ws + 477184;        // 128
    float* rmax  = ws + 477312;        // 128
    float* rsum  = ws + 477440;        // 128
    float* s_cat = ws + 477568;        // 128*1024   = 131072  (total ~2.44 MB)

    embed_kernel<<<BB, HD, 0, stream>>>(input, emb, emb_x);
    gates_kernel<<<dim3((4 * H2) / 16, BB / 16), 32, 0, stream>>>(
        emb_x, dh, w_ih, w_hh, b_ih, b_hh, gates);
    lstm_pw_kernel<<<(BB * H2 + 255) / 256, 256, 0, stream>>>(gates, cell, h_new, c_new);
    hW_kernel<<<BB, H2, 0, stream>>>(h_new, w_att1, b_att1, hW);
    att_score_kernel<<<dim3(TK / 16, BB), 32, 0, stream>>>(enc, w_att1, hW, w_att2, e_a);
    att_softmax_kernel<<<BB, 512, 0, stream>>>(e_a);
    context_kernel<<<BB, H2, 0, stream>>>(e_a, enc, ctx);
    scat_build_kernel<<<(BB * 2 * H2 + 255) / 256, 256, 0, stream>>>(h_new, ctx, s_cat);
    pgen_kernel<<<BB, 256, 0, stream>>>(ctx, h_new, emb_x, w_gen, b_gen, pgen);
    fc_kernel<<<VOC / 16, 256, 0, stream>>>(s_cat, w_fc, b_fc, out);
    rowstat_kernel<<<BB, 256, 0, stream>>>(out, rmax, rsum);
    {
        size_t n = (size_t)BB * VEXT;
        pvocab_kernel<<<(unsigned)((n + 255) / 256), 256, 0, stream>>>(out, rmax, rsum, pgen);
    }
    scatter_kernel<<<1, BB, 0, stream>>>(out, story, e_a, pgen);
    {
        size_t n = (size_t)BB * VEXT;
        log_kernel<<<(unsigned)((n + 255) / 256), 256, 0, stream>>>(out);
    }
}